// DMN_Encoder_58798102282929
// MI455X (gfx1250) — compile-verified
//
#include <hip/hip_runtime.h>
#include <hip/hip_bf16.h>
#include <math.h>

// Problem constants (match reference)
#define B_ 128
#define N_ 2048
#define D_ 128
#define HOPS 3
#define NCHUNK 8   // n-dimension split for the big o-pass (B*NCHUNK = 1024 blocks)

typedef __attribute__((ext_vector_type(2))) float v2f;
typedef __attribute__((ext_vector_type(8))) float v8f;

__device__ __forceinline__ float wave_reduce_sum(float v) {
#pragma unroll
    for (int off = 16; off > 0; off >>= 1)
        v += __shfl_xor(v, off, 32);
    return v;
}

// ---------------------------------------------------------------- copy e1 -> u
__global__ void k_copy(const float* __restrict__ src, float* __restrict__ dst, int n) {
    int i = blockIdx.x * blockDim.x + threadIdx.x;
    if (i < n) dst[i] = src[i];
}

// ------------------------------------------------ fw[b,n] = mask * (value . w_f)
// One wave32 per (b,n) row: lane loads float4 of the 128-float row, dot with w_f,
// wave reduce. Hops only change the scalar s[b], so this is computed once.
__global__ void k_fw(const float* __restrict__ value, const float* __restrict__ mask,
                     const float* __restrict__ attw, float* __restrict__ fw) {
    const int w = threadIdx.x >> 5, lane = threadIdx.x & 31;
    const long row = (long)blockIdx.x * 8 + w;              // row < B*N
    const float4 v  = ((const float4*)(value + row * D_))[lane];
    const float4 wf = ((const float4*)attw)[lane];          // w_f = attfc_w[0, 0:D]
    float dot = v.x * wf.x + v.y * wf.y + v.z * wf.z + v.w * wf.w;
    dot = wave_reduce_sum(dot);
    if (lane == 0) fw[row] = mask[row] * dot;
}

// ----------------------------------------------------------- s[b] = u[b] . w_u
__global__ void k_s(const float* __restrict__ u, const float* __restrict__ attw,
                    float* __restrict__ s) {
    const int w = threadIdx.x >> 5, lane = threadIdx.x & 31;
    const int b = blockIdx.x * 8 + w;                       // b < 128
    const float4 uv = ((const float4*)(u + b * D_))[lane];
    const float4 wu = ((const float4*)(attw + D_))[lane];   // w_u = attfc_w[0, D:2D]
    float dot = uv.x * wu.x + uv.y * wu.y + uv.z * wu.z + uv.w * wu.w;
    dot = wave_reduce_sum(dot);
    if (lane == 0) s[b] = dot;
}

// --------------------------- u_batch = relu(u @ linfc_w^T + bias) via f32 WMMA
// One wave per 16x16 C tile; K=128 in steps of 4 via V_WMMA_F32_16X16X4_F32.
// f32 A 16x4 layout: lanes 0-15 hold (K=0,K=1), lanes 16-31 hold (K=2,K=3).
// B[k,j] = linfc_w[j,k] (we need u @ W^T), same split across lane halves.
// C/D: VGPR r -> row (r + 8*hi), col laneN. Block=32 => EXEC all ones (required).
__global__ void k_linfc(const float* __restrict__ u, const float* __restrict__ W,
                        const float* __restrict__ bias, float* __restrict__ ub) {
    const int lane  = threadIdx.x;
    const int laneN = lane & 15;
    const int hi    = lane >> 4;
    const int rowBase = blockIdx.y * 16;
    const int colBase = blockIdx.x * 16;
    v8f c = {};
#if __has_builtin(__builtin_amdgcn_wmma_f32_16x16x4_f32)
    const float* arow = u + (rowBase + laneN) * D_;
    const float* brow = W + (colBase + laneN) * D_;
#pragma unroll 4
    for (int k0 = 0; k0 < D_; k0 += 4) {
        const int kk = k0 + 2 * hi;
        v2f a; a.x = arow[kk]; a.y = arow[kk + 1];
        v2f b; b.x = brow[kk]; b.y = brow[kk + 1];
        c = __builtin_amdgcn_wmma_f32_16x16x4_f32(
            /*neg_a=*/false, a, /*neg_b=*/false, b,
            /*c_mod=*/(short)0, c, /*reuse_a=*/false, /*reuse_b=*/false);
    }
#else
    // Scalar fallback with identical C layout (should not be taken on gfx1250).
#pragma unroll
    for (int r = 0; r < 8; ++r) {
        const int row = rowBase + r + 8 * hi;
        const int col = colBase + laneN;
        float acc = 0.f;
        for (int k = 0; k < D_; ++k) acc += u[row * D_ + k] * W[col * D_ + k];
        c[r] = acc;
    }
#endif
    const float bv = bias[colBase + laneN];
#pragma unroll
    for (int r = 0; r < 8; ++r) {
        const int row = rowBase + r + 8 * hi;
        const float val = c[r] + bv;
        ub[row * D_ + colBase + laneN] = val > 0.f ? val : 0.f;
    }
}

// --------------------- softmax stats per b: maxv[b], invd[b] = 1/(sum + 1e-5)
__global__ void k_stats(const float* __restrict__ fw, const float* __restrict__ mask,
                        const float* __restrict__ s, const float* __restrict__ attb,
                        float* __restrict__ maxv, float* __restrict__ invd) {
    __shared__ float red[256];
    const int b = blockIdx.x;
    const int t = threadIdx.x;
    const float sb = s[b] + attb[0];
    const float* fwr = fw + (long)b * N_;
    const float* mkr = mask + (long)b * N_;
    float att[8];
    float m = -1e30f;
#pragma unroll
    for (int i = 0; i < 8; ++i) {
        float a = fwr[t + 256 * i] + sb;
        a = a > 0.f ? a : 0.f;          // relu (reference applies relu before max)
        att[i] = a;
        m = fmaxf(m, a);
    }
    red[t] = m; __syncthreads();
    for (int off = 128; off > 0; off >>= 1) {
        if (t < off) red[t] = fmaxf(red[t], red[t + off]);
        __syncthreads();
    }
    const float M = red[0]; __syncthreads();
    float sum = 0.f;
#pragma unroll
    for (int i = 0; i < 8; ++i)
        sum += __expf(att[i] - M) * mkr[t + 256 * i];
    red[t] = sum; __syncthreads();
    for (int off = 128; off > 0; off >>= 1) {
        if (t < off) red[t] += red[t + off];
        __syncthreads();
    }
    if (t == 0) { maxv[b] = M; invd[b] = 1.f / (red[0] + 1e-5f); }
}

// --------------- big pass: partial o = sum_n q[b,n] * value[b,n,:] per chunk
// q = exp(relu(fw+s+c)-M) * mask / denom  (mask^2 == mask, binary).
// Wave-per-row, lane = d/4 (float4). mask==0 rows skipped (wave-uniform branch),
// cutting ~30% of the 128 MB features traffic. value is L2-resident (128<192MB).
__global__ void k_o(const float* __restrict__ value, const float* __restrict__ mask,
                    const float* __restrict__ fw, const float* __restrict__ s,
                    const float* __restrict__ attb, const float* __restrict__ maxv,
                    const float* __restrict__ invd, float* __restrict__ partials) {
    __shared__ float red[8 * D_];
    const int b     = blockIdx.y;
    const int chunk = blockIdx.x;
    const int w     = threadIdx.x >> 5;
    const int lane  = threadIdx.x & 31;
    const float sb  = s[b] + attb[0];
    const float M   = maxv[b];
    const float inv = invd[b];
    const long base = (long)b * N_;
    const int n0 = chunk * 256 + w * 32;      // 32 contiguous rows per wave
    const float4* vrow = (const float4*)(value + (base + n0) * D_);
    float4 acc = make_float4(0.f, 0.f, 0.f, 0.f);
    for (int i = 0; i < 32; ++i) {
        if (i + 2 < 32)
            __builtin_prefetch(&vrow[(i + 2) * 32 + lane], 0, 0);
        const int n = n0 + i;
        const float mk = mask[base + n];
        if (mk != 0.f) {
            float a = fw[base + n] + sb;
            a = a > 0.f ? a : 0.f;
            const float q = __expf(a - M) * mk * inv;
            const float4 v = vrow[i * 32 + lane];
            acc.x += q * v.x; acc.y += q * v.y;
            acc.z += q * v.z; acc.w += q * v.w;
        }
    }
    ((float4*)red)[w * 32 + lane] = acc;
    __syncthreads();
    const int t = threadIdx.x;
    if (t < D_) {
        float o = 0.f;
#pragma unroll
        for (int ww = 0; ww < 8; ++ww) o += red[ww * D_ + t];
        partials[((long)b * NCHUNK + chunk) * D_ + t] = o;
    }
}

// --------------------------------- u_new = u_batch + sum(chunk partials)
__global__ void k_update(const float* __restrict__ ub, const float* __restrict__ partials,
                         float* __restrict__ dst) {
    const int b = blockIdx.x, t = threadIdx.x;
    float o = 0.f;
#pragma unroll
    for (int c = 0; c < NCHUNK; ++c)
        o += partials[((long)b * NCHUNK + c) * D_ + t];
    dst[b * D_ + t] = ub[b * D_ + t] + o;
}

extern "C" void kernel_launch(void* const* d_in, const int* in_sizes, int n_in,
                              void* d_out, int out_size, void* d_ws, size_t ws_size,
                              hipStream_t stream) {
    const float* e1     = (const float*)d_in[0];   // (B,D)
    // d_in[1] rel_embeded  : unused by reference
    // d_in[2] nei_embeded_key : unused by reference
    const float* value  = (const float*)d_in[3];   // (B,N,D)
    const float* mask   = (const float*)d_in[4];   // (B,N)
    const float* linw   = (const float*)d_in[5];   // (D,D)
    const float* linb   = (const float*)d_in[6];   // (D,)
    const float* attw   = (const float*)d_in[7];   // (1,2D) -> w_f | w_u
    const float* attb   = (const float*)d_in[8];   // (1,)
    float* out = (float*)d_out;

    // Workspace layout (floats), ~1.7 MB total
    char* ws = (char*)d_ws;
    float* fw    = (float*)(ws);                               // B*N      = 1 MB
    float* s     = (float*)(ws + 1048576);                     // B
    float* maxv  = (float*)(ws + 1048576 + 512);               // B
    float* invd  = (float*)(ws + 1048576 + 1024);              // B
    float* u_cur = (float*)(ws + 1048576 + 1536);              // B*D
    float* ub    = (float*)(ws + 1048576 + 1536 + 65536);      // B*D
    float* part  = (float*)(ws + 1048576 + 1536 + 131072);     // B*NCHUNK*D = 512 KB

    k_copy<<<64, 256, 0, stream>>>(e1, u_cur, B_ * D_);
    k_fw<<<(B_ * N_) / 8, 256, 0, stream>>>(value, mask, attw, fw);

    for (int h = 0; h < HOPS; ++h) {
        k_linfc<<<dim3(D_ / 16, B_ / 16), 32, 0, stream>>>(u_cur, linw, linb, ub);
        k_s<<<B_ / 8, 256, 0, stream>>>(u_cur, attw, s);
        k_stats<<<B_, 256, 0, stream>>>(fw, mask, s, attb, maxv, invd);
        k_o<<<dim3(NCHUNK, B_), 256, 0, stream>>>(value, mask, fw, s, attb, maxv, invd, part);
        float* dst = (h == HOPS - 1) ? out : u_cur;
        k_update<<<B_, D_, 0, stream>>>(ub, part, dst);
    }
}